// PushingEnv_65635690218006
// MI455X (gfx1250) — compile-verified
//
#include <hip/hip_runtime.h>

// ---------------------------------------------------------------------------
// PushingEnv rollout: B independent 4-state scans over H=200 steps.
// Memory-bound (105 MB streamed once -> ~4.6us floor at 23.3 TB/s).
// Strategy: TDM (tensor_load_to_lds) DMAs 256-row x 25-col tiles of u_seq
// into LDS, double buffered, tracked with TENSORcnt; 256 threads each own one
// batch row and integrate from LDS (conflict-free: 25-dword row stride is odd).
// Reads use explicit address_space(3) volatile pointers built from LDS byte
// offsets -> guaranteed ds_load_b32, un-deletable by alias analysis.
// ---------------------------------------------------------------------------

#define H_STEPS 200
#define DT 0.01f
#define KSPR 100.0f
#define CDAMP 1.0f

#define ROWS 256                       // batch rows per workgroup (= blockDim.x)
#define CHUNK 25                       // timesteps per TDM chunk
#define NCHUNK (H_STEPS / CHUNK)       // 8 chunks
#define CHUNK_BYTES (ROWS * CHUNK * 4) // 25600 B per LDS buffer

typedef __attribute__((ext_vector_type(4))) unsigned int uint32x4;
typedef __attribute__((ext_vector_type(8))) int int32x8;
typedef __attribute__((ext_vector_type(4))) int int32x4;

// LDS (addrspace 3) volatile element: loads via this type must lower to ds_*.
typedef __attribute__((address_space(3))) volatile float lds_vf32;

// Compiler-only barrier: models "the TDM may have written LDS here".
__device__ __forceinline__ void lds_compiler_fence() {
  asm volatile("" ::: "memory");
}

// Issue one TDM 2D tile load: ROWS x CHUNK fp32 tile, row stride H, into LDS.
// Descriptor layout per CDNA5 ISA (D# groups 0/1).
// This toolchain exposes the 6-arg builtin:
//   (uint32x4 g0, int32x8 g1, int32x4 g2, int32x4 g3, int32x8 extra, i32 cpol)
__device__ __forceinline__ void tdm_load_chunk(const float* gbase, unsigned lds_off) {
  unsigned long long ga = (unsigned long long)(const void*)gbase;
  uint32x4 g0;
  g0.x = 1u;                                            // count=1 (valid), user mode
  g0.y = lds_off;                                       // lds_addr (bytes)
  g0.z = (unsigned)(ga & 0xFFFFFFFFull);                // global_addr[31:0]
  g0.w = (unsigned)((ga >> 32) & 0x01FFFFFFull)         // global_addr[56:32]
       | (2u << 30);                                    // type=2 ("image")
  int32x8 g1;
  g1[0] = (2 << 16);                                    // data_size=4B; mask=0; no pad/iter
  g1[1] = (CHUNK & 0xFFFF) << 16;                       // tensor_dim0 lo16 = 25
  g1[2] = (ROWS & 0xFFFF) << 16;                        // dim0 hi=0 | tensor_dim1 lo16 = 256
  g1[3] = (CHUNK & 0xFFFF) << 16;                       // dim1 hi=0 | tile_dim0 = 25
  g1[4] = ROWS;                                         // tile_dim1 = 256 | tile_dim2 = 0
  g1[5] = H_STEPS;                                      // tensor_dim0_stride lo32 = 200
  g1[6] = 0;                                            // stride0 hi | stride1 lo
  g1[7] = 0;                                            // stride1 hi
  int32x4 z4 = {0, 0, 0, 0};                            // groups 2/3: 2-D tile, dims unused
  int32x8 z8 = {0, 0, 0, 0, 0, 0, 0, 0};                // extra descriptor group: unused
  __builtin_amdgcn_tensor_load_to_lds(g0, g1, z4, z4, z8, 0); // cpol = default
}

__global__ __launch_bounds__(ROWS) void pushing_env_scan_kernel(
    const float* __restrict__ u_seq,   // [B, H] row-major
    const float* __restrict__ x1_0,    // [B]
    const float* __restrict__ v1_0,    // [B]
    const float* __restrict__ x2_0,    // [B]
    const float* __restrict__ v2_0,    // [B]
    const float* __restrict__ goal,    // [1]
    float* __restrict__ out,           // [3*B]: loss | x1 | x2
    int B)
{
  __shared__ float smem[2][ROWS * CHUNK];   // 50 KB double buffer, LDS offset 0

  // Escape smem's address so alias analysis must assume opaque agents (the
  // TDM / later asm memory clobbers) can write it.
  {
    const void* esc = (const void*)&smem[0][0];
    asm volatile("" :: "v"(esc) : "memory");
  }

  const int tid  = threadIdx.x;
  const int row0 = blockIdx.x * ROWS;
  const int b    = row0 + tid;

  float x1 = x1_0[b];
  float v1 = v1_0[b];
  float x2 = x2_0[b];
  float v2 = v2_0[b];

  const bool leader = (tid < 32);                    // wave 0 drives the TDM
  const float* tile_base = u_seq + (size_t)row0 * H_STEPS;

  if (leader) {                                      // prologue: 2 chunks in flight
    tdm_load_chunk(tile_base + 0 * CHUNK, 0u);
    tdm_load_chunk(tile_base + 1 * CHUNK, (unsigned)CHUNK_BYTES);
  }

  for (int c = 0; c < NCHUNK; ++c) {
    if (leader) {
      // Loads complete in order per wave: <=1 outstanding guarantees chunk c done.
      if (c < NCHUNK - 1) __builtin_amdgcn_s_wait_tensorcnt(1);
      else                __builtin_amdgcn_s_wait_tensorcnt(0);
    }
    __syncthreads();        // chunk c visible to all waves
    lds_compiler_fence();   // TDM wrote smem behind the compiler's back

    // AS3 pointer built from the raw LDS byte offset (smem is at offset 0,
    // same convention as the TDM descriptors). Volatile AS3 load -> ds_load_b32.
    const unsigned row_off = (unsigned)((c & 1) * CHUNK_BYTES + tid * (CHUNK * 4));
    lds_vf32* urow = (lds_vf32*)row_off;
    #pragma unroll
    for (int t = 0; t < CHUNK; ++t) {
      float u       = urow[t];
      float pen     = x1 + 1.0f - x2;                // spring engages on touch
      float contact = fmaxf(pen, 0.0f);
      float Fc      = -KSPR * contact;
      float F1      = u + Fc - CDAMP * v1;
      float F2      = -Fc - CDAMP * v2;
      v1 = fmaf(F1, DT, v1);                         // semi-implicit: v first,
      v2 = fmaf(F2, DT, v2);
      x1 = fmaf(v1, DT, x1);                         // then x with updated v
      x2 = fmaf(v2, DT, x2);
    }

    lds_compiler_fence();   // keep the reads above the buffer-reuse point
    __syncthreads();        // all waves done with buf[c&1]
    if (leader && (c + 2) < NCHUNK) {                // refill the freed buffer
      tdm_load_chunk(tile_base + (c + 2) * CHUNK, (unsigned)((c & 1) * CHUNK_BYTES));
    }
  }

  const float g = goal[0];
  const float d = x2 - g;
  out[b]         = d * d;   // loss
  out[B + b]     = x1;
  out[2 * B + b] = x2;
}

extern "C" void kernel_launch(void* const* d_in, const int* in_sizes, int n_in,
                              void* d_out, int out_size, void* d_ws, size_t ws_size,
                              hipStream_t stream) {
  const float* u_seq = (const float*)d_in[0];
  const float* x1_0  = (const float*)d_in[1];
  const float* v1_0  = (const float*)d_in[2];
  const float* x2_0  = (const float*)d_in[3];
  const float* v2_0  = (const float*)d_in[4];
  const float* goal  = (const float*)d_in[5];
  float* out = (float*)d_out;

  const int B = in_sizes[1];          // 131072; divisible by ROWS
  const int nblocks = B / ROWS;       // 512 workgroups of 8 wave32s

  pushing_env_scan_kernel<<<nblocks, ROWS, 0, stream>>>(
      u_seq, x1_0, v1_0, x2_0, v2_0, goal, out, B);
}